// GraphResidualNetwork_70514773065745
// MI455X (gfx1250) — compile-verified
//
#include <hip/hip_runtime.h>
#include <hip/hip_bf16.h>
#include <cmath>

#define GN_NODES 100000
#define GN_EDGES 1600000
#define GD 64
#define GN_LAYERS 8
#define GN_GRAPHS 512
#define GOUT_DIM 32
#define GALPHA 0.5f

typedef float v2f __attribute__((ext_vector_type(2)));
typedef float v8f __attribute__((ext_vector_type(8)));

// ---------------------------------------------------------------------------
// Kernel 1: x0 = x @ W_in + b_in          (FP32 WMMA 16x16x4, 1 wave = 16 rows)
// ---------------------------------------------------------------------------
__global__ __launch_bounds__(256) void input_proj_kernel(
    const float* __restrict__ x, const float* __restrict__ W,
    const float* __restrict__ b, float* __restrict__ x0) {
  const int gwave = (blockIdx.x * 256 + threadIdx.x) >> 5;
  if (gwave * 16 >= GN_NODES) return;            // wave-uniform exit
  const int lane = threadIdx.x & 31;
  const int row  = lane & 15;                    // A row / B,D column
  const int koff = (lane >> 4) << 1;             // lanes 16-31 hold K+2,K+3
  const int grow = gwave * 16 + row;

  v8f acc[4] = {};                               // 4 column tiles of 16
  const float* arow = x + (size_t)grow * GD;
#pragma unroll
  for (int k = 0; k < 16; ++k) {
    const int kk = 4 * k + koff;
    v2f a = *(const v2f*)(arow + kk);            // A frag: rows of x
#pragma unroll
    for (int nt = 0; nt < 4; ++nt) {
      const float* wp = W + (size_t)kk * GD + nt * 16 + row;
      v2f bf; bf.x = wp[0]; bf.y = wp[GD];       // B frag: W[kk][col], W[kk+1][col]
      acc[nt] = __builtin_amdgcn_wmma_f32_16x16x4_f32(
          false, a, false, bf, (short)0, acc[nt], false, false);
    }
  }
  // D layout: lanes 0-15 -> M = v, lanes 16-31 -> M = v+8; col = lane&15
  const int mbase = (lane >> 4) << 3;
#pragma unroll
  for (int nt = 0; nt < 4; ++nt) {
#pragma unroll
    for (int v = 0; v < 8; ++v) {
      const int c = nt * 16 + row;
      x0[(size_t)(gwave * 16 + mbase + v) * GD + c] = acc[nt][v] + b[c];
    }
  }
}

// ---------------------------------------------------------------------------
// Kernel 2: edge scatter  agg[dst] += w * h[src]   (16 lanes per edge, float4)
// ---------------------------------------------------------------------------
__global__ __launch_bounds__(256) void scatter_kernel(
    const float* __restrict__ h, const int* __restrict__ src,
    const int* __restrict__ dst, const float* __restrict__ ew,
    float* __restrict__ agg) {
  const int t = blockIdx.x * 256 + threadIdx.x;
  const int e = t >> 4;
  const int q = t & 15;
  if (e >= GN_EDGES) return;
  const int s = src[e];
  const int d = dst[e];
  const float w = ew[e];
  const float4 v = *(const float4*)(h + (size_t)s * GD + q * 4);
  float* op = agg + (size_t)d * GD + q * 4;
  atomicAdd(op + 0, w * v.x);
  atomicAdd(op + 1, w * v.y);
  atomicAdd(op + 2, w * v.z);
  atomicAdd(op + 3, w * v.w);
}

// ---------------------------------------------------------------------------
// Kernel 3: fused GCN2 layer.
//   hmid = (1-a)*agg + a*x0
//   h_out = relu((1-beta)*hmid + beta*(hmid @ W) + h_in)
// ---------------------------------------------------------------------------
__global__ __launch_bounds__(256) void gcn2_layer_kernel(
    const float* __restrict__ agg, const float* __restrict__ x0,
    const float* __restrict__ h_in, const float* __restrict__ W,
    float* __restrict__ h_out, float beta) {
  const int gwave = (blockIdx.x * 256 + threadIdx.x) >> 5;
  if (gwave * 16 >= GN_NODES) return;
  const int lane = threadIdx.x & 31;
  const int row  = lane & 15;
  const int koff = (lane >> 4) << 1;
  const int grow = gwave * 16 + row;
  const float ia = 1.0f - GALPHA;

  v8f acc[4] = {};
  const float* ag_row = agg + (size_t)grow * GD;
  const float* x0_row = x0 + (size_t)grow * GD;
#pragma unroll
  for (int k = 0; k < 16; ++k) {
    const int kk = 4 * k + koff;
    v2f g = *(const v2f*)(ag_row + kk);
    v2f z = *(const v2f*)(x0_row + kk);
    v2f a;
    a.x = ia * g.x + GALPHA * z.x;
    a.y = ia * g.y + GALPHA * z.y;
#pragma unroll
    for (int nt = 0; nt < 4; ++nt) {
      const float* wp = W + (size_t)kk * GD + nt * 16 + row;
      v2f bf; bf.x = wp[0]; bf.y = wp[GD];
      acc[nt] = __builtin_amdgcn_wmma_f32_16x16x4_f32(
          false, a, false, bf, (short)0, acc[nt], false, false);
    }
  }
  const int mbase = (lane >> 4) << 3;
  const float ib = 1.0f - beta;
#pragma unroll
  for (int nt = 0; nt < 4; ++nt) {
#pragma unroll
    for (int v = 0; v < 8; ++v) {
      const size_t idx = (size_t)(gwave * 16 + mbase + v) * GD + nt * 16 + row;
      const float hm  = ia * agg[idx] + GALPHA * x0[idx];
      const float val = ib * hm + beta * acc[nt][v] + h_in[idx];
      h_out[idx] = fmaxf(val, 0.0f);
    }
  }
}

// ---------------------------------------------------------------------------
// Kernel 4: per-graph sum pooling (atomic) + node counts
// ---------------------------------------------------------------------------
__global__ __launch_bounds__(256) void pool_scatter_kernel(
    const float* __restrict__ h, const int* __restrict__ batch,
    float* __restrict__ pooled, float* __restrict__ cnts) {
  const int t = blockIdx.x * 256 + threadIdx.x;
  const int n = t >> 4;
  const int q = t & 15;
  if (n >= GN_NODES) return;
  const int g = batch[n];
  const float4 v = *(const float4*)(h + (size_t)n * GD + q * 4);
  float* op = pooled + (size_t)g * GD + q * 4;
  atomicAdd(op + 0, v.x);
  atomicAdd(op + 1, v.y);
  atomicAdd(op + 2, v.z);
  atomicAdd(op + 3, v.w);
  if (q == 0) atomicAdd(cnts + g, 1.0f);
}

// ---------------------------------------------------------------------------
// Kernel 5: out = (pooled / max(cnt,1)) @ W_out + b_out        (512x64x32)
// ---------------------------------------------------------------------------
__global__ __launch_bounds__(256) void out_proj_kernel(
    const float* __restrict__ pooled, const float* __restrict__ cnts,
    const float* __restrict__ W_out, const float* __restrict__ b_out,
    float* __restrict__ out) {
  const int t = blockIdx.x * 256 + threadIdx.x;
  if (t >= GN_GRAPHS * GOUT_DIM) return;
  const int g = t >> 5;           // OUT_DIM == 32
  const int o = t & 31;
  const float inv = 1.0f / fmaxf(cnts[g], 1.0f);
  float s = 0.0f;
#pragma unroll
  for (int d = 0; d < GD; ++d)
    s += pooled[(size_t)g * GD + d] * W_out[(size_t)d * GOUT_DIM + o];
  out[t] = s * inv + b_out[o];
}

// ---------------------------------------------------------------------------
extern "C" void kernel_launch(void* const* d_in, const int* in_sizes, int n_in,
                              void* d_out, int out_size, void* d_ws, size_t ws_size,
                              hipStream_t stream) {
  const float* x     = (const float*)d_in[0];
  const int*   eidx  = (const int*)d_in[1];
  const float* ew    = (const float*)d_in[2];
  const int*   batch = (const int*)d_in[3];
  const float* W_in  = (const float*)d_in[4];
  const float* b_in  = (const float*)d_in[5];
  const float* Ws    = (const float*)d_in[6];
  const float* W_out = (const float*)d_in[7];
  const float* b_out = (const float*)d_in[8];
  float* out = (float*)d_out;

  const size_t ND = (size_t)GN_NODES * GD;
  float* ws     = (float*)d_ws;
  float* x0     = ws;                 // [N, 64]
  float* agg    = ws + ND;            // [N, 64]
  float* hA     = ws + 2 * ND;        // [N, 64]
  float* hB     = ws + 3 * ND;        // [N, 64]
  float* pooled = ws + 4 * ND;        // [G, 64]
  float* cnts   = pooled + (size_t)GN_GRAPHS * GD;  // [G]

  const int* src = eidx;
  const int* dst = eidx + GN_EDGES;

  const int ntiles = GN_NODES / 16;               // 6250 (exact)
  dim3 gemm_grid((ntiles + 7) / 8);               // 8 waves per 256-thread block

  // x0 = x @ W_in + b_in ; h starts as x0
  input_proj_kernel<<<gemm_grid, 256, 0, stream>>>(x, W_in, b_in, x0);

  const float* h_in = x0;
  float* h_out = hA;
  for (int i = 0; i < GN_LAYERS; ++i) {
    hipMemsetAsync(agg, 0, ND * sizeof(float), stream);
    scatter_kernel<<<(GN_EDGES * 16) / 256, 256, 0, stream>>>(h_in, src, dst, ew, agg);
    const float beta = logf(1.0f / (float)(i + 1) + 1.0f);
    gcn2_layer_kernel<<<gemm_grid, 256, 0, stream>>>(
        agg, x0, h_in, Ws + (size_t)i * GD * GD, h_out, beta);
    h_in  = h_out;
    h_out = (h_out == hA) ? hB : hA;
  }

  hipMemsetAsync(pooled, 0,
                 ((size_t)GN_GRAPHS * GD + GN_GRAPHS) * sizeof(float), stream);
  pool_scatter_kernel<<<(GN_NODES * 16 + 255) / 256, 256, 0, stream>>>(
      h_in, batch, pooled, cnts);
  out_proj_kernel<<<(GN_GRAPHS * GOUT_DIM + 255) / 256, 256, 0, stream>>>(
      pooled, cnts, W_out, b_out, out);
}